// Mamba2Block_84318797955519
// MI455X (gfx1250) — compile-verified
//
#include <hip/hip_runtime.h>
#include <cmath>

typedef __attribute__((ext_vector_type(2))) float v2f;
typedef __attribute__((ext_vector_type(8))) float v8f;

#define WMMA_F32(a, b, c) \
  __builtin_amdgcn_wmma_f32_16x16x4_f32(false, (a), false, (b), (short)0, (c), false, false)

// Async global->LDS copy (CDNA5, ASYNCcnt-tracked). GV mode: per-lane 64-bit
// global address, per-lane LDS byte offset in the VDST slot.
__device__ __forceinline__ void async_g2l_b128(uint32_t lds_byte_off, const float* gp) {
  asm volatile("global_load_async_to_lds_b128 %0, %1, off"
               :: "v"(lds_byte_off), "v"(gp) : "memory");
}
__device__ __forceinline__ void wait_async0() {
  asm volatile("s_wait_asynccnt 0x0" ::: "memory");
}
__device__ __forceinline__ uint32_t lds_off(const void* p) {
  return (uint32_t)(uintptr_t)p;   // generic ptr low 32 bits = LDS byte offset
}

// ---------------------------------------------------------------------------
// fp32 WMMA GEMM:  C[M,N] = A[M,K] @ W[N,K]^T   (row-major fp32)
// Block = 64x64 tile, 4 waves x (16M x 64N). K-slabs of 32 staged in LDS by
// the async global->LDS engine, double-buffered: slab ks+1 is issued before
// the 32 WMMAs of slab ks, then a single s_wait_asynccnt 0 + barrier
// publishes it. Row pad 36 floats: conflict-free ds_load_b64 (36*r mod 64
// distinct for r<16) and 144B row stride keeps b128 LDS writes 16B-aligned.
// ---------------------------------------------------------------------------
__global__ void __launch_bounds__(128)
gemm_f32_wmma(const float* __restrict__ A, const float* __restrict__ W,
              float* __restrict__ C, int M, int N, int K) {
  __shared__ float As[2][64][36];
  __shared__ float Ws[2][64][36];

  const int tid  = threadIdx.x;
  const int lane = tid & 31;
  const int wave = tid >> 5;
  const int tiles_n = N >> 6;
  const int m0 = (blockIdx.x / tiles_n) << 6;
  const int n0 = (blockIdx.x % tiles_n) << 6;

  // cooperative slab copy: 2 threads per row, 16 floats (4 x b128) each
  const int lrow = tid >> 1;
  const int lcol = (tid & 1) << 4;
  const float* ga = A + (size_t)(m0 + lrow) * K + lcol;
  const float* gw = W + (size_t)(n0 + lrow) * K + lcol;

  // prologue: async-copy slab 0 into buffer 0
  #pragma unroll
  for (int i = 0; i < 4; ++i) {
    async_g2l_b128(lds_off(&As[0][lrow][lcol + i * 4]), ga + i * 4);
    async_g2l_b128(lds_off(&Ws[0][lrow][lcol + i * 4]), gw + i * 4);
  }
  wait_async0();
  __syncthreads();

  const int r    = lane & 15;
  const int kh   = (lane >> 4) << 1;        // K-half select: 0 or 2
  const int arow = (wave << 4) + r;         // this wave's 16 M-rows

  v8f acc0 = {}, acc1 = {}, acc2 = {}, acc3 = {};
  const int nslab = K >> 5;
  for (int ks = 0; ks < nslab; ++ks) {
    const int buf = ks & 1;
    if (ks + 1 < nslab) {
      // async-copy next slab into buf^1 (its readers finished at the barrier
      // that ended iteration ks-1; disjoint from buf being read below)
      const float* ga2 = ga + (ks + 1) * 32;
      const float* gw2 = gw + (ks + 1) * 32;
      #pragma unroll
      for (int i = 0; i < 4; ++i) {
        async_g2l_b128(lds_off(&As[buf ^ 1][lrow][lcol + i * 4]), ga2 + i * 4);
        async_g2l_b128(lds_off(&Ws[buf ^ 1][lrow][lcol + i * 4]), gw2 + i * 4);
      }
    }
    #pragma unroll
    for (int kk = 0; kk < 32; kk += 4) {    // 8 x (ds_load + 4 WMMA)
      v2f a  = *(const v2f*)&As[buf][arow  ][kk + kh];
      v2f b0 = *(const v2f*)&Ws[buf][r     ][kk + kh];
      v2f b1 = *(const v2f*)&Ws[buf][r + 16][kk + kh];
      v2f b2 = *(const v2f*)&Ws[buf][r + 32][kk + kh];
      v2f b3 = *(const v2f*)&Ws[buf][r + 48][kk + kh];
      acc0 = WMMA_F32(a, b0, acc0);
      acc1 = WMMA_F32(a, b1, acc1);
      acc2 = WMMA_F32(a, b2, acc2);
      acc3 = WMMA_F32(a, b3, acc3);
    }
    if (ks + 1 < nslab) wait_async0();      // own async writes done (after math)
    __syncthreads();                        // all waves' writes visible
  }

  float* cp = C + (size_t)(m0 + (wave << 4) + ((lane >> 4) << 3)) * N + n0 + r;
  #pragma unroll
  for (int v = 0; v < 8; ++v) {
    cp[(size_t)v * N +  0] = acc0[v];
    cp[(size_t)v * N + 16] = acc1[v];
    cp[(size_t)v * N + 32] = acc2[v];
    cp[(size_t)v * N + 48] = acc3[v];
  }
}

// ---------------------------------------------------------------------------
// Causal depthwise conv (k=4) + bias + SiLU, both directions.
// xz: [b][l][2048] (ch 0..1023 = pre-conv xc, 1024..2047 = z).
// xc: [dir][b][l][1024]; bwd reads xz time-flipped.
// ---------------------------------------------------------------------------
__global__ void __launch_bounds__(256)
conv_silu_kernel(const float* __restrict__ xz, const float* __restrict__ cw,
                 const float* __restrict__ cb, float* __restrict__ xc) {
  const int L = 1024;
  const size_t idx = (size_t)blockIdx.x * blockDim.x + threadIdx.x;
  const int d   = (int)(idx & 1023);
  const int l   = (int)((idx >> 10) & 1023);
  const int bb  = (int)((idx >> 20) & 1);
  const int dir = (int)(idx >> 21);

  float acc = cb[d];
  #pragma unroll
  for (int j = 0; j < 4; ++j) {
    const int t = l - 3 + j;
    if (t >= 0) {
      const int tt = dir ? (L - 1 - t) : t;
      acc = fmaf(cw[d * 4 + j], xz[((size_t)bb * L + tt) * 2048 + d], acc);
    }
  }
  xc[idx] = acc / (1.0f + expf(-acc));
}

// ---------------------------------------------------------------------------
// Chunked selective scan. dA is scalar per (channel,t), so the recurrence
// splits exactly:  s_t = p_t * s_chunkstart + s_loc,t  with p_t = prod(dA).
// Pass A: per 64-step chunk from zero state; y_loc to ybuf; cumulative p_t
//         OVERWRITES the consumed delta slot of proj; chunk-end state to sloc.
// ---------------------------------------------------------------------------
__global__ void __launch_bounds__(256)
scanA_kernel(const float* __restrict__ xc, float* __restrict__ proj,
             const float* __restrict__ A_log, float* __restrict__ ybuf,
             float* __restrict__ sloc) {
  const int L = 1024, Dm = 1024, NP = 1152, T = 64, NC = 16;
  const int pb = blockIdx.z;                 // dir*2 + b
  const int ck = blockIdx.y;                 // chunk 0..15
  const int d  = blockIdx.x * 256 + threadIdx.x;

  __shared__ float Bs[2][64], Cs[2][64];

  const float Ac = -expf(A_log[d]);
  float*       pp = proj + (size_t)pb * L * NP;
  const float* xp = xc   + (size_t)pb * L * Dm;
  float*       yp = ybuf + (size_t)pb * L * Dm;

  float s[64];
  #pragma unroll
  for (int n = 0; n < 64; ++n) s[n] = 0.0f;
  float p = 1.0f;

  const int t0 = ck * T;
  for (int tt = 0; tt < T; ++tt) {
    const int t = t0 + tt;
    const int buf = tt & 1;
    if (threadIdx.x < 128) {
      const int n = threadIdx.x & 63;
      float v = pp[(size_t)t * NP + 1024 + threadIdx.x];
      if (threadIdx.x < 64) Bs[buf][n] = v; else Cs[buf][n] = v;
    }
    __syncthreads();

    const float up = pp[(size_t)t * NP + d];
    const float dt = (up > 20.0f) ? up : log1pf(expf(up));   // softplus
    const float u  = xp[(size_t)t * Dm + d];
    const float dA = expf(dt * Ac);
    const float du = dt * u;
    p *= dA;

    float y = 0.0f;
    #pragma unroll
    for (int n = 0; n < 64; ++n) {
      s[n] = fmaf(dA, s[n], du * Bs[buf][n]);
      y    = fmaf(Cs[buf][n], s[n], y);
    }
    pp[(size_t)t * NP + d] = p;              // delta slot -> cumulative product
    yp[(size_t)t * Dm + d] = y;              // local (uncorrected) output
  }

  float* sp = sloc + (((size_t)pb * NC + ck) * Dm + d) * 64;
  #pragma unroll
  for (int n = 0; n < 64; ++n) sp[n] = s[n];
}

// Pass B: tiny serial sweep over 16 chunks, parallel over (pb,d,n) = 262144.
// sloc is rewritten in place with the chunk-START state.
__global__ void __launch_bounds__(256)
scanB_kernel(const float* __restrict__ proj, float* __restrict__ sloc) {
  const int L = 1024, Dm = 1024, NP = 1152, T = 64, NC = 16;
  const size_t idx = (size_t)blockIdx.x * 256 + threadIdx.x;
  const int n  = (int)(idx & 63);
  const int d  = (int)((idx >> 6) & 1023);
  const int pb = (int)(idx >> 16);

  const float* pp = proj + (size_t)pb * L * NP + d;
  float*       sp = sloc + ((size_t)pb * NC * Dm + (size_t)d) * 64 + n;

  float run = 0.0f;
  for (int c = 0; c < NC; ++c) {
    const float P = pp[(size_t)(c * T + T - 1) * NP];        // full-chunk prod
    float* sc = sp + (size_t)c * Dm * 64;
    const float tmp = *sc;
    *sc = run;                                               // start state
    run = fmaf(P, run, tmp);
  }
}

// Pass C: chunk-parallel correction + D skip + SiLU(z) gating, in place over
// ybuf. Start states live in VGPRs; C_t broadcast via double-buffered LDS.
__global__ void __launch_bounds__(256)
scanC_kernel(const float* __restrict__ proj, const float* __restrict__ sstart,
             float* __restrict__ ybuf, const float* __restrict__ xc,
             const float* __restrict__ xz, const float* __restrict__ Dvec) {
  const int L = 1024, Dm = 1024, NP = 1152, T = 64, NC = 16;
  const int pb  = blockIdx.z;
  const int ck  = blockIdx.y;
  const int d   = blockIdx.x * 256 + threadIdx.x;
  const int dir = pb >> 1, bb = pb & 1;

  __shared__ float Cs[2][64];

  const float Dd = Dvec[d];
  const float* pp = proj + (size_t)pb * L * NP;
  const float* xp = xc   + (size_t)pb * L * Dm;
  const float* zp = xz   + (size_t)bb * L * 2048 + 1024;
  float*       yp = ybuf + (size_t)pb * L * Dm;

  float s[64];
  const float* sp = sstart + (((size_t)pb * NC + ck) * Dm + d) * 64;
  #pragma unroll
  for (int n = 0; n < 64; ++n) s[n] = sp[n];

  const int t0 = ck * T;
  for (int tt = 0; tt < T; ++tt) {
    const int t = t0 + tt;
    const int buf = tt & 1;
    if (threadIdx.x < 64)
      Cs[buf][threadIdx.x] = pp[(size_t)t * NP + 1088 + threadIdx.x];
    __syncthreads();

    const float p = pp[(size_t)t * NP + d];                  // cumulative prod
    float dot = 0.0f;
    #pragma unroll
    for (int n = 0; n < 64; ++n) dot = fmaf(Cs[buf][n], s[n], dot);

    const float u = xp[(size_t)t * Dm + d];
    const float y = yp[(size_t)t * Dm + d] + p * dot + Dd * u;
    const int tz  = dir ? (L - 1 - t) : t;
    const float z = zp[(size_t)tz * 2048 + d];
    yp[(size_t)t * Dm + d] = y * (z / (1.0f + expf(-z)));
  }
}

// out[b][l][c] = 0.5 * (od_fwd[b][l][c] + od_bwd[b][L-1-l][c]) ; od: [dir][b][l][512]
__global__ void __launch_bounds__(256)
combine_kernel(const float* __restrict__ od, float* __restrict__ out) {
  const int L = 1024, C = 512;
  const size_t idx = (size_t)blockIdx.x * blockDim.x + threadIdx.x;
  const int c  = (int)(idx & 511);
  const int l  = (int)((idx >> 9) & 1023);
  const int bb = (int)(idx >> 19);
  const size_t bwd = ((size_t)(2 + bb) * L + (L - 1 - l)) * C + c;
  out[idx] = 0.5f * (od[idx] + od[bwd]);
}

extern "C" void kernel_launch(void* const* d_in, const int* in_sizes, int n_in,
                              void* d_out, int out_size, void* d_ws, size_t ws_size,
                              hipStream_t stream) {
  const float* x      = (const float*)d_in[0];
  const float* W_in   = (const float*)d_in[1];
  const float* conv_w = (const float*)d_in[2];
  const float* conv_b = (const float*)d_in[3];
  const float* W_xprj = (const float*)d_in[4];
  const float* A_log  = (const float*)d_in[5];
  const float* Dv     = (const float*)d_in[6];
  const float* W_out  = (const float*)d_in[7];
  float* out = (float*)d_out;

  float* ws    = (float*)d_ws;
  float* xzw   = ws;                                   // 2*1024*2048
  float* xcw   = xzw   + (size_t)2 * 1024 * 2048;      // 4*1024*1024
  float* projw = xcw   + (size_t)4 * 1024 * 1024;      // 4*1024*1152
  float* ybw   = projw + (size_t)4 * 1024 * 1152;      // 4*1024*1024 (y_loc -> y_gated)
  float* slocw = ybw   + (size_t)4 * 1024 * 1024;      // 4*16*1024*64
  float* odw   = slocw + (size_t)4 * 16 * 1024 * 64;   // 4*1024*512

  // 1) xz = x @ W_in^T — once; both directions index it (flipped for bwd)
  gemm_f32_wmma<<<(2048 / 64) * (2048 / 64), 128, 0, stream>>>(x, W_in, xzw, 2048, 2048, 512);
  // 2) causal conv + bias + SiLU per direction
  conv_silu_kernel<<<(4 * 1024 * 1024) / 256, 256, 0, stream>>>(xzw, conv_w, conv_b, xcw);
  // 3) proj = xc @ W_xproj^T, directions stacked in M
  gemm_f32_wmma<<<(4096 / 64) * (1152 / 64), 128, 0, stream>>>(xcw, W_xprj, projw, 4096, 1152, 1024);
  // 4) chunked selective scan (A: local chunks, B: chunk stitch, C: correction+gate)
  scanA_kernel<<<dim3(4, 16, 4), 256, 0, stream>>>(xcw, projw, A_log, ybw, slocw);
  scanB_kernel<<<(4 * 1024 * 64) / 256, 256, 0, stream>>>(projw, slocw);
  scanC_kernel<<<dim3(4, 16, 4), 256, 0, stream>>>(projw, slocw, ybw, xcw, xzw, Dv);
  // 5) od = y_gated @ W_out^T
  gemm_f32_wmma<<<(4096 / 64) * (512 / 64), 128, 0, stream>>>(ybw, W_out, odw, 4096, 512, 1024);
  // 6) average fwd with time-flipped bwd
  combine_kernel<<<(2 * 1024 * 512) / 256, 256, 0, stream>>>(odw, out);
}